// LaplacianLoss_18537078849648
// MI455X (gfx1250) — compile-verified
//
#include <hip/hip_runtime.h>

// LaplacianLoss for MI455X (gfx1250, wave32).
// Memory-bound gather kernel: one thread per (b,n) node.
//  - B*N = 1,600,000 nodes, E = 10 neighbor slots (-1 = invalid), D = 3.
//  - A_list stream (64 MB int32) dominates HBM traffic; coord gathers are
//    L2-resident (1.2 MB per batch slab, 192 MB L2). Roofline ~6 us.
//  - Block reduction uses v_wmma_f32_16x16x4_f32 (A = 64 partials, B = ones)
//    for an exact, deterministic 64->16 row-sum, finished with ds_bpermute
//    shuffles. Two-pass reduction via d_ws keeps fp order fixed every replay.

typedef __attribute__((ext_vector_type(2))) float v2f;
typedef __attribute__((ext_vector_type(8))) float v8f;

#define BB     16
#define NN     100000
#define EE     10
#define TPB    256
#define NODES  (BB * NN)          // 1,600,000
#define NBLK   (NODES / TPB)      // 6250, exact (no tail)

__global__ __launch_bounds__(TPB) void lap_partial_kernel(
    const float* __restrict__ c1,
    const float* __restrict__ c2,
    const int*   __restrict__ A,
    float*       __restrict__ partial)
{
    __shared__ float red[TPB];

    const int tid = threadIdx.x;
    const int gid = blockIdx.x * TPB + tid;        // node id, always < NODES
    const int b   = gid / NN;
    const int n   = gid - b * NN;

    const float* cb1  = c1 + (size_t)b * NN * 3;
    const float* cb2  = c2 + (size_t)b * NN * 3;
    const int*   Arow = A + (size_t)gid * EE;

    // Start pulling this node's own coordinates while we chase neighbors.
    __builtin_prefetch(cb1 + 3 * (size_t)n, 0, 3);   // -> global_prefetch_b8
    __builtin_prefetch(cb2 + 3 * (size_t)n, 0, 3);

    float s1x = 0.f, s1y = 0.f, s1z = 0.f;
    float s2x = 0.f, s2y = 0.f, s2z = 0.f;
    float cnt = 0.f;

#pragma unroll
    for (int e = 0; e < EE; ++e) {
        const int idx = Arow[e];
        if (idx >= 0) {
            const float* p1 = cb1 + 3 * (size_t)idx;
            const float* p2 = cb2 + 3 * (size_t)idx;
            s1x += p1[0]; s1y += p1[1]; s1z += p1[2];
            s2x += p2[0]; s2y += p2[1]; s2z += p2[2];
            cnt += 1.0f;
        }
    }

    // Mirror the reference exactly: cent = sum/cnt per coord set, lap = c - cent.
    const float l1x = cb1[3 * n + 0] - s1x / cnt;
    const float l1y = cb1[3 * n + 1] - s1y / cnt;
    const float l1z = cb1[3 * n + 2] - s1z / cnt;
    const float l2x = cb2[3 * n + 0] - s2x / cnt;
    const float l2y = cb2[3 * n + 1] - s2y / cnt;
    const float l2z = cb2[3 * n + 2] - s2z / cnt;

    const float dx = l2x - l1x;
    const float dy = l2y - l1y;
    const float dz = l2z - l1z;

    red[tid] = dx * dx + dy * dy + dz * dz;
    __syncthreads();

    // Wave 0 (32 lanes, EXEC all ones within the wave) reduces 256 partials.
    if (tid < 32) {
        float a0 = 0.f, a1 = 0.f;
#pragma unroll
        for (int i = 0; i < 4; ++i) a0 += red[tid * 8 + i];
#pragma unroll
        for (int i = 0; i < 4; ++i) a1 += red[tid * 8 + 4 + i];

        // 64 values = the full 16x4 f32 A-matrix (2 VGPRs/lane, wave32).
        v2f Am; Am.x = a0;  Am.y = a1;
        v2f Bm; Bm.x = 1.f; Bm.y = 1.f;   // 4x16 all-ones B
        v8f Cm = {};
        // D[m][n] = sum_k A[m][k]  -> 16 row sums, replicated across columns.
        v8f Dm = __builtin_amdgcn_wmma_f32_16x16x4_f32(
            false, Am, false, Bm, (short)0, Cm, false, false);

        float s = Dm[0] + Dm[1] + Dm[2] + Dm[3] + Dm[4] + Dm[5] + Dm[6] + Dm[7];
        // lanes 0-15 hold rows 0-7 sum, lanes 16-31 hold rows 8-15 sum.
        float tot = s + __shfl_xor(s, 16, 32);
        if (tid == 0) partial[blockIdx.x] = tot;
    }
}

__global__ __launch_bounds__(TPB) void lap_final_kernel(
    const float* __restrict__ partial,
    float*       __restrict__ out)
{
    __shared__ float red[TPB];

    float local = 0.f;
    for (int i = threadIdx.x; i < NBLK; i += TPB) local += partial[i];
    red[threadIdx.x] = local;
    __syncthreads();

#pragma unroll
    for (int s = TPB / 2; s > 0; s >>= 1) {
        if (threadIdx.x < s) red[threadIdx.x] += red[threadIdx.x + s];
        __syncthreads();
    }

    if (threadIdx.x == 0) {
        // loss = mean((lap2-lap1)^2) * N = sum / (B*N*D) * N
        const float mean = red[0] / (float)(NODES * 3);   // B*N*D = 4,800,000
        out[0] = mean * (float)NN;
    }
}

extern "C" void kernel_launch(void* const* d_in, const int* in_sizes, int n_in,
                              void* d_out, int out_size, void* d_ws, size_t ws_size,
                              hipStream_t stream)
{
    const float* c1 = (const float*)d_in[0];   // coord1 [B,N,D] f32
    const float* c2 = (const float*)d_in[1];   // coord2 [B,N,D] f32
    const int*   A  = (const int*)d_in[2];     // A_list [B,N,E] int32 (-1 invalid)
    float* out      = (float*)d_out;           // scalar loss
    float* partial  = (float*)d_ws;            // NBLK block partials (25 KB)

    lap_partial_kernel<<<NBLK, TPB, 0, stream>>>(c1, c2, A, partial);
    lap_final_kernel<<<1, TPB, 0, stream>>>(partial, out);
}